// V1Column_42442866819225
// MI455X (gfx1250) — compile-verified
//
#include <hip/hip_runtime.h>
#include <stdint.h>

typedef float v2f __attribute__((ext_vector_type(2)));
typedef float v8f __attribute__((ext_vector_type(8)));

// ---------------------------------------------------------------------------
// Non-returning f32 atomic add -> global_atomic_add_f32 (no CAS loop)
// ---------------------------------------------------------------------------
__device__ __forceinline__ void atomicAddF32(float* p, float v) {
  __hip_atomic_fetch_add(p, v, __ATOMIC_RELAXED, __HIP_MEMORY_SCOPE_AGENT);
}

// ---------------------------------------------------------------------------
// 1) Zero the workspace (acc buffer + spike bitmask)
// ---------------------------------------------------------------------------
__global__ void zero_ws_kernel(uint32_t* __restrict__ ws, long long nwords) {
  long long i = (long long)blockIdx.x * blockDim.x + threadIdx.x;
  long long stride = (long long)gridDim.x * blockDim.x;
  for (; i < nwords; i += stride) ws[i] = 0u;
}

// ---------------------------------------------------------------------------
// 2) Pack spikes into a bitmask: mask[b*WPB + j/32] bit (j%32) = spike[b,j]!=0
// ---------------------------------------------------------------------------
__global__ void pack_spikes_kernel(const float* __restrict__ rec, int rec_total,
                                   const int* __restrict__ npost_p, int out_rows,
                                   uint32_t* __restrict__ mask) {
  int i = blockIdx.x * blockDim.x + threadIdx.x;
  if (i >= rec_total) return;
  int npost = *npost_p;
  int B = out_rows / npost;
  int NPRE = rec_total / B;
  int b = i / NPRE;
  int j = i - b * NPRE;
  int WPB = (NPRE + 31) >> 5;
  if (rec[i] != 0.0f)
    atomicOr(&mask[b * WPB + (j >> 5)], 1u << (j & 31));
}

// ---------------------------------------------------------------------------
// 3) Gated scatter: acc[b, post, type] += w  for firing (b, synapse) pairs.
//    Spike bitmask staged in LDS; weight/syn_id loads deferred behind the
//    spike test (~96% of synapses skip them); 1 atomic per firing event.
// ---------------------------------------------------------------------------
__global__ __launch_bounds__(256)
void scatter_kernel(const int2* __restrict__ idx,
                    const int* __restrict__ syn_ids,
                    const float* __restrict__ wvals, int n_syn,
                    const uint32_t* __restrict__ mask_g,
                    float* __restrict__ acc,
                    const int* __restrict__ npost_p,
                    int rec_total, int out_rows, int ntypes) {
  extern __shared__ uint32_t smask[];
  int npost = *npost_p;
  int B = out_rows / npost;
  int NPRE = rec_total / B;
  int WPB = (NPRE + 31) >> 5;
  int mwords = B * WPB;
  for (int i = threadIdx.x; i < mwords; i += blockDim.x) smask[i] = mask_g[i];
  __syncthreads();

  long long s = (long long)blockIdx.x * blockDim.x + threadIdx.x;
  long long stride = (long long)gridDim.x * blockDim.x;
  for (; s < n_syn; s += stride) {
    // keep the 8B/synapse index stream ahead of us (-> global_prefetch_b8)
    __builtin_prefetch(&idx[s + 4 * stride], 0, 0);
    int2 pp = idx[s];                // .x = post, .y = pre
    int post = pp.x;
    int pre = pp.y;
    int wofs = pre >> 5;
    uint32_t bit = 1u << (pre & 31);

    // Which batches fired for this pre-neuron? (pure LDS, no global traffic)
    uint32_t hits = 0u;
    for (int b = 0; b < B; ++b)
      if (smask[b * WPB + wofs] & bit) hits |= (1u << b);

    if (hits) {                       // ~4% of synapses reach here
      float w = wvals[s];
      int t = syn_ids[s];
      for (int b = 0; b < B; ++b)
        if (hits & (1u << b))
          atomicAddF32(&acc[((long long)b * npost + post) * ntypes + t], w);
    }
  }
}

// ---------------------------------------------------------------------------
// 4) out[row, 0:5] = acc[row, 0:NT] @ basis[NT, 5] via V_WMMA_F32_16X16X4_F32.
//    NT compile-time -> fully unrolled 5-step K chain (s_clause'd b64 loads
//    overlapping the WMMA chain); B fragments hoisted out of the tile loop;
//    scalar (readfirstlane) tile-loop control so EXEC stays all-1s; full
//    tiles take an unguarded store path (1 address + immediate offsets).
// ---------------------------------------------------------------------------
template <int NT>
__global__ __launch_bounds__(256)
void gemm_basis_fixed(const float* __restrict__ acc,
                      const float* __restrict__ basis,
                      float* __restrict__ out, int nrows) {
  constexpr int KSTEPS = (NT + 3) / 4;
  int lane = threadIdx.x & 31;
  int m = lane & 15;
  int hi = lane >> 4;                  // 0: K pair {0,1}, 1: K pair {2,3}

  // Wave-uniform loop control, forced scalar.
  int waveId = __builtin_amdgcn_readfirstlane(
      blockIdx.x * (blockDim.x >> 5) + (threadIdx.x >> 5));
  int wavesTotal = __builtin_amdgcn_readfirstlane(
      (gridDim.x * blockDim.x) >> 5);

  // --- B fragments: depend only on lane, load once, reuse for every tile ---
  int n = (m < 5) ? m : 0;             // clamped, always-in-range address
  v2f bfrag[KSTEPS];
#pragma unroll
  for (int kk = 0; kk < KSTEPS; ++kk) {
    int col = kk * 4 + hi * 2;
    float bx = basis[col * 5 + n];           // unconditional load
    float by = basis[(col + 1) * 5 + n];     // unconditional load
    bfrag[kk].x = (m < 5) ? bx : 0.0f;       // v_cndmask, no branch
    bfrag[kk].y = (m < 5) ? by : 0.0f;
  }

  int ntiles = (nrows + 15) >> 4;
  for (int tile = waveId; tile < ntiles; tile += wavesTotal) {
    int R0 = tile << 4;
    int row = R0 + m;
    if (row >= nrows) row = nrows - 1;       // clamp; tail stores are guarded
    const float* arow = acc + (long long)row * NT;

    // Issue all A loads first so they overlap the WMMA chain.
    v2f afrag[KSTEPS];
#pragma unroll
    for (int kk = 0; kk < KSTEPS; ++kk)
      afrag[kk] = *(const v2f*)(arow + kk * 4 + hi * 2);  // global_load_b64

    v8f c = {0.f, 0.f, 0.f, 0.f, 0.f, 0.f, 0.f, 0.f};
#pragma unroll
    for (int kk = 0; kk < KSTEPS; ++kk)
      c = __builtin_amdgcn_wmma_f32_16x16x4_f32(
          false, afrag[kk], false, bfrag[kk], (short)0, c, false, false);

    // C/D layout: VGPR j holds (row R0 + 8*hi + j, col m). Only cols 0-4 real.
    float* op = out + (long long)(R0 + hi * 8) * 5 + m;  // row stride 5 floats
    if (R0 + 16 <= nrows) {                  // full tile: unguarded stores
      if (m < 5) {
#pragma unroll
        for (int j = 0; j < 8; ++j) op[5 * j] = c[j];  // offset:20*j
      }
    } else {                                 // (at most one) tail tile
      if (m < 5) {
        int rbase = R0 + hi * 8;
#pragma unroll
        for (int j = 0; j < 8; ++j)
          if (rbase + j < nrows) op[5 * j] = c[j];
      }
    }
  }
}

// Trivial scalar fallback for unexpected ntypes (clean codegen, no WMMA).
__global__ __launch_bounds__(256)
void gemm_basis_scalar(const float* __restrict__ acc,
                       const float* __restrict__ basis,
                       float* __restrict__ out,
                       int nrows, int ntypes) {
  int r = blockIdx.x * blockDim.x + threadIdx.x;
  if (r >= nrows) return;
  float s0 = 0.f, s1 = 0.f, s2 = 0.f, s3 = 0.f, s4 = 0.f;
  const float* arow = acc + (long long)r * ntypes;
  for (int t = 0; t < ntypes; ++t) {
    float a = arow[t];
    const float* br = basis + t * 5;
    s0 += a * br[0]; s1 += a * br[1]; s2 += a * br[2];
    s3 += a * br[3]; s4 += a * br[4];
  }
  float* orow = out + (long long)r * 5;
  orow[0] = s0; orow[1] = s1; orow[2] = s2; orow[3] = s3; orow[4] = s4;
}

// ---------------------------------------------------------------------------
// Launch: zero -> pack -> scatter -> wmma-gemm, all on `stream`.
// ---------------------------------------------------------------------------
extern "C" void kernel_launch(void* const* d_in, const int* in_sizes, int n_in,
                              void* d_out, int out_size, void* d_ws, size_t ws_size,
                              hipStream_t stream) {
  const float*    rec     = (const float*)d_in[0];     // [B, N_PRE]
  const float*    wvals   = (const float*)d_in[1];     // [N_SYN]
  const float*    basis   = (const float*)d_in[2];     // [N_TYPES, 5]
  const int2*     idx     = (const int2*)d_in[3];      // [N_SYN, 2] (post, pre)
  const int*      syn_ids = (const int*)d_in[4];       // [N_SYN]
  const int*      npost_p = (const int*)d_in[5];       // scalar n_post
  float*          out     = (float*)d_out;             // [B*n_post, 5]

  int rec_total = in_sizes[0];
  int n_syn     = in_sizes[1];
  int ntypes    = in_sizes[2] / 5;     // N_BASIS = 5 per reference
  int out_rows  = out_size / 5;        // B * n_post

  long long acc_elems  = (long long)out_rows * ntypes;             // f32
  long long mask_words = (long long)(rec_total + 31) / 32 + 32;    // upper bound
  float*    acc  = (float*)d_ws;
  uint32_t* mask = (uint32_t*)((char*)d_ws + acc_elems * sizeof(float));

  // 1) zero acc + mask region
  long long zwords = acc_elems + mask_words;
  int zblocks = (int)((zwords + 255) / 256);
  zero_ws_kernel<<<zblocks, 256, 0, stream>>>((uint32_t*)d_ws, zwords);

  // 2) pack spike bitmask
  int pblocks = (rec_total + 255) / 256;
  pack_spikes_kernel<<<pblocks, 256, 0, stream>>>(rec, rec_total, npost_p,
                                                  out_rows, mask);

  // 3) gated scatter into acc
  size_t smem = (size_t)mask_words * sizeof(uint32_t);
  scatter_kernel<<<2048, 256, smem, stream>>>(idx, syn_ids, wvals, n_syn, mask,
                                              acc, npost_p, rec_total, out_rows,
                                              ntypes);

  // 4) WMMA GEMM: out = acc @ basis
  if (ntypes == 20) {
    int tiles = (out_rows + 15) / 16;
    int gblocks = (tiles + 7) / 8;    // 8 waves (tiles) per 256-thread block
    gemm_basis_fixed<20><<<gblocks, 256, 0, stream>>>(acc, basis, out, out_rows);
  } else {
    int gblocks = (out_rows + 255) / 256;
    gemm_basis_scalar<<<gblocks, 256, 0, stream>>>(acc, basis, out, out_rows,
                                                   ntypes);
  }
}